// WCE_label_correction_noise_4552665333886
// MI455X (gfx1250) — compile-verified
//
#include <hip/hip_runtime.h>
#include <math.h>

// WCE with label-noise transition matrices.
// Streams: pred, targets, T[:,0], T[:,1], noisy_mask (reads) + p1 (write).
// ~402 MB @ 23.3 TB/s => ~17us; purely HBM-bound. No matrix contraction
// exists, so no WMMA; optimal path is vectorized global loads with NT hints.

#define WCE_EPS 1e-5f

constexpr int      TPB  = 256;               // 8 wave32 per block
constexpr int      NBLK = 2048;              // partial-sum blocks
constexpr unsigned HW   = 512u * 512u;       // 262144 = 2^18
constexpr unsigned NTOT = 64u * HW;          // 16777216 = 2^24
constexpr unsigned NV4  = NTOT / 4u;         // 4194304 vec4 elements

typedef float f32x4 __attribute__((ext_vector_type(4)));

__global__ __launch_bounds__(TPB) void wce_main_kernel(
    const float* __restrict__ pred,
    const float* __restrict__ targ,
    const float* __restrict__ T,
    const float* __restrict__ mask,
    float* __restrict__ p1_out,       // d_out + 1 (4B-aligned only)
    float* __restrict__ partials)     // d_ws: [NBLK] s1 then [NBLK] s2
{
  const unsigned stride = (unsigned)NBLK * TPB;
  float s1 = 0.0f;
  float s2 = 0.0f;

  for (unsigned v4 = blockIdx.x * TPB + threadIdx.x; v4 < NV4; v4 += stride) {
    const unsigned i  = v4 * 4u;
    const unsigned b  = i >> 18;             // i / HW
    const unsigned hw = i & (HW - 1u);       // i % HW
    const size_t   t0off = ((size_t)b << 19) + hw;   // b*2*HW + hw

    // Prefetch next grid-stride iteration (speculative; OOB silently dropped).
    __builtin_prefetch(pred + i + (size_t)stride * 4u, 0, 0);
    __builtin_prefetch(targ + i + (size_t)stride * 4u, 0, 0);
    __builtin_prefetch(mask + i + (size_t)stride * 4u, 0, 0);

    // 16B-aligned streaming loads, non-temporal (no reuse -> keep L2 clean).
    const f32x4 p  = __builtin_nontemporal_load((const f32x4*)(pred + i));
    const f32x4 tg = __builtin_nontemporal_load((const f32x4*)(targ + i));
    const f32x4 m  = __builtin_nontemporal_load((const f32x4*)(mask + i));
    const f32x4 t0 = __builtin_nontemporal_load((const f32x4*)(T + t0off));
    const f32x4 t1 = __builtin_nontemporal_load((const f32x4*)(T + t0off + HW));

    f32x4 p1v;
#pragma unroll
    for (int k = 0; k < 4; ++k) {
      const float omp = 1.0f - p[k];
      float p0 = t0[k] * omp + (1.0f - t1[k]) * p[k];
      float p1 = (1.0f - t0[k]) * omp + t1[k] * p[k];
      p0 = fminf(fmaxf(p0, 0.0f), 1.0f);
      p1 = fminf(fmaxf(p1, 0.0f), 1.0f);
      p1v[k] = p1;
      const float temp = tg[k] * logf(p1 + WCE_EPS)
                       + (1.0f - tg[k]) * logf(p0 + WCE_EPS);
      s1 += temp * m[k];
      s2 += m[k];
    }

    // d_out+1 is only 4B-aligned -> scalar b32 NT stores (still coalesced).
#pragma unroll
    for (int k = 0; k < 4; ++k)
      __builtin_nontemporal_store(p1v[k], p1_out + i + k);
  }

  // wave32 tree reduction (gfx1250 is wave32-only).
#pragma unroll
  for (int off = 16; off > 0; off >>= 1) {
    s1 += __shfl_down(s1, off, 32);
    s2 += __shfl_down(s2, off, 32);
  }

  __shared__ float l1[TPB / 32];
  __shared__ float l2[TPB / 32];
  const int lane = threadIdx.x & 31;
  const int wid  = threadIdx.x >> 5;
  if (lane == 0) { l1[wid] = s1; l2[wid] = s2; }
  __syncthreads();

  if (threadIdx.x == 0) {
    float a = 0.0f, c = 0.0f;
#pragma unroll
    for (int k = 0; k < TPB / 32; ++k) { a += l1[k]; c += l2[k]; }
    partials[blockIdx.x]        = a;   // sum(temp * mask) partial
    partials[NBLK + blockIdx.x] = c;   // sum(mask) partial
  }
}

// Deterministic single-block finalize: loss = -sum1/sum2 -> d_out[0].
__global__ __launch_bounds__(TPB) void wce_final_kernel(
    const float* __restrict__ partials, float* __restrict__ out)
{
  float s1 = 0.0f, s2 = 0.0f;
  for (unsigned j = threadIdx.x; j < (unsigned)NBLK; j += TPB) {
    s1 += partials[j];
    s2 += partials[NBLK + j];
  }
#pragma unroll
  for (int off = 16; off > 0; off >>= 1) {
    s1 += __shfl_down(s1, off, 32);
    s2 += __shfl_down(s2, off, 32);
  }
  __shared__ float l1[TPB / 32];
  __shared__ float l2[TPB / 32];
  const int lane = threadIdx.x & 31;
  const int wid  = threadIdx.x >> 5;
  if (lane == 0) { l1[wid] = s1; l2[wid] = s2; }
  __syncthreads();
  if (threadIdx.x == 0) {
    float a = 0.0f, c = 0.0f;
#pragma unroll
    for (int k = 0; k < TPB / 32; ++k) { a += l1[k]; c += l2[k]; }
    out[0] = -(a / c);
  }
}

extern "C" void kernel_launch(void* const* d_in, const int* in_sizes, int n_in,
                              void* d_out, int out_size, void* d_ws, size_t ws_size,
                              hipStream_t stream) {
  (void)in_sizes; (void)n_in; (void)out_size; (void)ws_size;
  const float* pred = (const float*)d_in[0];
  const float* targ = (const float*)d_in[1];
  const float* T    = (const float*)d_in[2];
  const float* mask = (const float*)d_in[3];
  float* out      = (float*)d_out;          // [0] = loss, [1..NTOT] = p1
  float* partials = (float*)d_ws;           // 2 * NBLK floats (16 KB)

  wce_main_kernel<<<NBLK, TPB, 0, stream>>>(pred, targ, T, mask, out + 1, partials);
  wce_final_kernel<<<1, TPB, 0, stream>>>(partials, out);
}